// GNN_6433861009920
// MI455X (gfx1250) — compile-verified
//
#include <hip/hip_runtime.h>

typedef float v2f __attribute__((ext_vector_type(2)));
typedef float v8f __attribute__((ext_vector_type(8)));

#define IN_DIM 128
#define HID    16
#define OUT_D  2

__device__ __forceinline__ int gtid() { return blockIdx.x * blockDim.x + threadIdx.x; }

// ---------------------------------------------------------------- utilities
__global__ void zero_f32(float* __restrict__ p, long long n) {
  long long i = (long long)blockIdx.x * blockDim.x + threadIdx.x;
  if (i < n) p[i] = 0.0f;
}

// ---------------------------------------------------------------- degree / norm
__global__ void deg_count(const int* __restrict__ dst, int E, unsigned* __restrict__ deg) {
  int e = gtid();
  if (e < E) atomicAdd(&deg[dst[e]], 1u);
}

__global__ void dinv_kernel(const unsigned* __restrict__ deg, float* __restrict__ dinv, int n) {
  int i = gtid();
  if (i < n) dinv[i] = rsqrtf((float)deg[i] + 1.0f);  // +1 for self-loop; always > 0
}

// ---------------------------------------------------------------- layer-1 GEMM: h1 = x @ W1 (WMMA f32 16x16x4)
// One wave computes a 16(M) x 16(N) f32 tile, K=128 in 32 steps of 4.
// A frag (16x4):   lane holds A[M = lane%16][K = kbase + 2*(lane/16) + j], j=0,1
// B frag (4x16):   lane holds B[K = kbase + j + 2*(lane/16)][N = lane%16], j=0,1
// C/D (16x16 f32): vgpr v, lane -> D[row = v + 8*(lane/16)][col = lane%16]
__global__ void gemm1_wmma(const float* __restrict__ x, const float* __restrict__ W,
                           float* __restrict__ h, int n) {
  int wave = (blockIdx.x * blockDim.x + threadIdx.x) >> 5;
  int lane = threadIdx.x & 31;
  int row0 = wave * 16;
  if (row0 >= n) return;               // wave-uniform: EXEC stays all-ones below
  int half = lane >> 4;                // 0 or 1
  int lrow = lane & 15;

  if (row0 + 16 <= n) {                // full tile: matrix path
    v8f c = {0.f, 0.f, 0.f, 0.f, 0.f, 0.f, 0.f, 0.f};
    const float* arow = x + (size_t)(row0 + lrow) * IN_DIM + 2 * half;
#pragma unroll
    for (int k = 0; k < IN_DIM; k += 4) {
      v2f a, b;
      a.x = arow[k];
      a.y = arow[k + 1];
      b.x = W[(k + 2 * half + 0) * HID + lrow];
      b.y = W[(k + 2 * half + 1) * HID + lrow];
      c = __builtin_amdgcn_wmma_f32_16x16x4_f32(false, a, false, b, (short)0, c,
                                                false, false);
    }
    float* out = h + (size_t)row0 * HID;
#pragma unroll
    for (int v = 0; v < 8; ++v)
      out[(size_t)(v + 8 * half) * HID + lrow] = c[v];
  } else {                             // ragged tail: scalar fallback (unused when n % 16 == 0)
    int r = row0 + lane;
    if (lane < 16 && r < n) {
      for (int f = 0; f < HID; ++f) {
        float acc = 0.f;
        for (int k = 0; k < IN_DIM; ++k) acc += x[(size_t)r * IN_DIM + k] * W[k * HID + f];
        h[(size_t)r * HID + f] = acc;
      }
    }
  }
}

// ---------------------------------------------------------------- layer-1 scatter: agg1[dst] += h1[src]*norm
// 4 threads per edge, each owning 4 contiguous features (float4 gather, 4 f32 L2 atomics)
__global__ void scatter1_kernel(const int* __restrict__ src, const int* __restrict__ dst,
                                const float* __restrict__ dinv, const float* __restrict__ h1,
                                float* __restrict__ agg1, int E) {
  int tid = gtid();
  int e = tid >> 2;
  if (e >= E) return;
  int g = (tid & 3) << 2;
  int s = src[e], d = dst[e];
  float w = dinv[s] * dinv[d];
  float4 v = *(const float4*)(h1 + (size_t)s * HID + g);
  float* o = agg1 + (size_t)d * HID + g;
  atomicAdd(o + 0, v.x * w);
  atomicAdd(o + 1, v.y * w);
  atomicAdd(o + 2, v.z * w);
  atomicAdd(o + 3, v.w * w);
}

// ---------------------------------------------------------------- self-loop + bias + ReLU (in place on agg1)
__global__ void relu_bias_kernel(float* __restrict__ agg1, const float* __restrict__ h1,
                                 const float* __restrict__ dinv, const float* __restrict__ b1,
                                 int n) {
  int tid = gtid();
  if (tid >= n * HID) return;
  int i = tid >> 4, f = tid & 15;
  float di = dinv[i];
  float v = agg1[tid] + h1[tid] * di * di + b1[f];
  agg1[tid] = v > 0.f ? v : 0.f;
}

// ---------------------------------------------------------------- layer-2 GEMM: h3 = h2 @ W2 (16x2, scalar)
__global__ void gemm2_kernel(const float* __restrict__ h2, const float* __restrict__ W2,
                             float* __restrict__ h3, int n) {
  int i = gtid();
  if (i >= n) return;
  const float* row = h2 + (size_t)i * HID;
  float o0 = 0.f, o1 = 0.f;
#pragma unroll
  for (int f = 0; f < HID; ++f) {
    float v = row[f];
    o0 += v * W2[f * 2 + 0];
    o1 += v * W2[f * 2 + 1];
  }
  h3[(size_t)i * 2 + 0] = o0;
  h3[(size_t)i * 2 + 1] = o1;
}

// ---------------------------------------------------------------- layer-2 scatter
__global__ void scatter2_kernel(const int* __restrict__ src, const int* __restrict__ dst,
                                const float* __restrict__ dinv, const float* __restrict__ h3,
                                float* __restrict__ agg2, int E) {
  int e = gtid();
  if (e >= E) return;
  int s = src[e], d = dst[e];
  float w = dinv[s] * dinv[d];
  atomicAdd(&agg2[(size_t)d * 2 + 0], h3[(size_t)s * 2 + 0] * w);
  atomicAdd(&agg2[(size_t)d * 2 + 1], h3[(size_t)s * 2 + 1] * w);
}

// ---------------------------------------------------------------- final: + self-loop + b2
__global__ void final_kernel(const float* __restrict__ agg2, const float* __restrict__ h3,
                             const float* __restrict__ dinv, const float* __restrict__ b2,
                             float* __restrict__ out, int n) {
  int tid = gtid();
  if (tid >= n * OUT_D) return;
  int i = tid >> 1, f = tid & 1;
  float di = dinv[i];
  out[tid] = agg2[tid] + h3[tid] * di * di + b2[f];
}

// ---------------------------------------------------------------- launcher
extern "C" void kernel_launch(void* const* d_in, const int* in_sizes, int n_in,
                              void* d_out, int out_size, void* d_ws, size_t ws_size,
                              hipStream_t stream) {
  (void)n_in; (void)out_size; (void)ws_size;
  const float* x   = (const float*)d_in[0];
  const int*   ei  = (const int*)d_in[1];     // [2, E] int32, row-major
  const float* W1  = (const float*)d_in[2];
  const float* b1  = (const float*)d_in[3];
  const float* W2  = (const float*)d_in[4];
  const float* b2  = (const float*)d_in[5];

  const int N = in_sizes[0] / IN_DIM;
  const int E = in_sizes[1] / 2;
  const int* src = ei;
  const int* dst = ei + E;

  // workspace layout (256B aligned)
  auto al = [](size_t v) { return (v + 255) & ~(size_t)255; };
  char* ws = (char*)d_ws;
  size_t a = 0;
  unsigned* deg  = (unsigned*)(ws + a); a = al(a + (size_t)N * 4);
  float*    dinv = (float*)(ws + a);    a = al(a + (size_t)N * 4);
  float*    h1   = (float*)(ws + a);    a = al(a + (size_t)N * HID * 4);
  float*    agg1 = (float*)(ws + a);    a = al(a + (size_t)N * HID * 4);
  float*    h3   = (float*)(ws + a);    a = al(a + (size_t)N * OUT_D * 4);
  float*    agg2 = (float*)(ws + a);    a = al(a + (size_t)N * OUT_D * 4);
  const size_t total_bytes = a;

  auto cdiv = [](long long x_, long long y_) { return (int)((x_ + y_ - 1) / y_); };
  const int B = 256;

  // zero deg/agg buffers every call (graph-replay safe); h1/h3 fully overwritten anyway
  long long zelems = (long long)(total_bytes / 4);
  zero_f32<<<cdiv(zelems, B), B, 0, stream>>>((float*)d_ws, zelems);

  deg_count<<<cdiv(E, B), B, 0, stream>>>(dst, E, deg);
  dinv_kernel<<<cdiv(N, B), B, 0, stream>>>(deg, dinv, N);

  long long waves = (N + 15) / 16;
  gemm1_wmma<<<cdiv(waves * 32, B), B, 0, stream>>>(x, W1, h1, N);

  scatter1_kernel<<<cdiv((long long)E * 4, B), B, 0, stream>>>(src, dst, dinv, h1, agg1, E);
  relu_bias_kernel<<<cdiv((long long)N * HID, B), B, 0, stream>>>(agg1, h1, dinv, b1, N);

  gemm2_kernel<<<cdiv(N, B), B, 0, stream>>>(agg1, W2, h3, N);
  scatter2_kernel<<<cdiv(E, B), B, 0, stream>>>(src, dst, dinv, h3, agg2, E);
  final_kernel<<<cdiv((long long)N * OUT_D, B), B, 0, stream>>>(agg2, h3, dinv, b2,
                                                                (float*)d_out, N);
}